// GCN3D_Jan14_66116726555401
// MI455X (gfx1250) — compile-verified
//
#include <hip/hip_runtime.h>
#include <math.h>

typedef float v2f __attribute__((ext_vector_type(2)));
typedef float v8f __attribute__((ext_vector_type(8)));

#define DEV_B   8
#define DEV_N   4096
#define DEV_C   512
#define DEV_M   (DEV_B * DEV_C)     // 4096 pooled clusters
#define NTOT    (DEV_B * DEV_N)     // 32768 nodes
#define F256    256

// ---------------------------------------------------------------------------
// Generic fill
// ---------------------------------------------------------------------------
__global__ __launch_bounds__(256) void fill_f32(float* p, float v, long long n) {
    long long i  = (long long)blockIdx.x * 256 + threadIdx.x;
    long long st = (long long)gridDim.x * 256;
    for (; i < n; i += st) p[i] = v;
}

// ---------------------------------------------------------------------------
// Degrees:  deg[d] = 1 + indeg(d);  dis = rsqrt(deg)
// ---------------------------------------------------------------------------
__global__ __launch_bounds__(256) void deg_accum(const int* __restrict__ dst,
                                                 float* __restrict__ deg, int E) {
    int i = blockIdx.x * 256 + threadIdx.x;
    if (i < E) atomicAdd(&deg[dst[i]], 1.0f);
}
__global__ __launch_bounds__(256) void deg_final(float* dis, int n) {
    int i = blockIdx.x * 256 + threadIdx.x;
    if (i < n) dis[i] = rsqrtf(dis[i] + 1.0f);   // +1 self loop
}

// ---------------------------------------------------------------------------
// Narrow WMMA fp32 GEMM (wave tile 16x16, block tile 64x32) — used only for
// the N=32 contraction (w_t1). C = act( A @ B + bias ).
// ---------------------------------------------------------------------------
__global__ __launch_bounds__(256)
void wmma_gemm(const float* __restrict__ A, const float* __restrict__ B,
               const float* __restrict__ bias, float* __restrict__ C,
               int M, int N, int K, int act) {
    const int lane = threadIdx.x & 31;
    const int wid  = threadIdx.x >> 5;
    const int half = lane >> 4;
    const int lid  = lane & 15;
    const int m0   = blockIdx.y * 64 + (wid & 3) * 16;
    const int n0   = blockIdx.x * 32 + (wid >> 2) * 16;

    const float* Arow = A + (long long)(m0 + lid) * K;
    const float* Bcol = B + (n0 + lid);
    const int kb = half * 2;

    v8f acc = {};
#pragma unroll 2
    for (int k0 = 0; k0 < K; k0 += 4) {
        v2f a, b;
        a.x = Arow[k0 + kb];
        a.y = Arow[k0 + kb + 1];
        b.x = Bcol[(long long)(k0 + kb) * N];
        b.y = Bcol[(long long)(k0 + kb + 1) * N];
        acc = __builtin_amdgcn_wmma_f32_16x16x4_f32(
            false, a, false, b, (short)0, acc, false, false);
    }

    const float bsv = bias ? bias[n0 + lid] : 0.0f;
#pragma unroll
    for (int r = 0; r < 8; ++r) {
        int   row = m0 + r + 8 * half;
        float v   = acc[r] + bsv;
        if (act == 1) v = v > 0.0f ? v : expm1f(v);
        C[(long long)row * N + (n0 + lid)] = v;
    }
}

// ---------------------------------------------------------------------------
// Wide WMMA fp32 GEMM:  C = act( rscale[row] * (A @ B) + bias[col] )
// Wave tile 16(M) x 64(N) with 4 accumulators sharing one A fragment
// (4x register reuse of A; 2.25 vmem per wmma). Block = 8 waves arranged
// 4(M) x 2(N) -> block tile 64 x 128. Requires M%64==0, N%128==0, K%4==0.
// Batched via blockIdx.z with element strides sA/sB/sC.
// Prefetches B eight K-rows ahead and A one 256B group ahead
// (global_prefetch_b8 path on gfx1250).
// ---------------------------------------------------------------------------
__global__ __launch_bounds__(256)
void wmma_gemm_w(const float* __restrict__ A, const float* __restrict__ B,
                 const float* __restrict__ bias, const float* __restrict__ rscale,
                 float* __restrict__ C, int M, int N, int K, int act,
                 long long sA, long long sB, long long sC, int rsStride) {
    const float* Ab = A + (long long)blockIdx.z * sA;
    const float* Bb = B + (long long)blockIdx.z * sB;
    float*       Cb = C + (long long)blockIdx.z * sC;

    const int lane = threadIdx.x & 31;
    const int wid  = threadIdx.x >> 5;
    const int half = lane >> 4;          // 0: lanes 0-15, 1: lanes 16-31
    const int lid  = lane & 15;
    const int m0   = blockIdx.y * 64 + (wid & 3) * 16;
    const int n0   = blockIdx.x * 128 + (wid >> 2) * 64;

    const float* Arow = Ab + (long long)(m0 + lid) * K;
    const int kb = half * 2;             // K sub-pair handled by this lane half

    v8f acc0 = {}, acc1 = {}, acc2 = {}, acc3 = {};
#pragma unroll 2
    for (int k0 = 0; k0 < K; k0 += 4) {
        v2f a;
        a.x = Arow[k0 + kb];
        a.y = Arow[k0 + kb + 1];

        const float* Brow0 = Bb + (long long)(k0 + kb) * N + n0 + lid;
        const float* Brow1 = Brow0 + N;

        // software prefetch: B eight K-rows ahead, A one 64-float group ahead
        if (k0 + 8 < K) __builtin_prefetch(Brow0 + (long long)8 * N, 0, 1);
        if (((k0 & 63) == 0) && (k0 + 64 < K))
            __builtin_prefetch(Arow + k0 + 64, 0, 1);

        v2f b0, b1, b2, b3;
        b0.x = Brow0[0];  b0.y = Brow1[0];
        b1.x = Brow0[16]; b1.y = Brow1[16];
        b2.x = Brow0[32]; b2.y = Brow1[32];
        b3.x = Brow0[48]; b3.y = Brow1[48];

        acc0 = __builtin_amdgcn_wmma_f32_16x16x4_f32(false, a, false, b0, (short)0, acc0, false, false);
        acc1 = __builtin_amdgcn_wmma_f32_16x16x4_f32(false, a, false, b1, (short)0, acc1, false, false);
        acc2 = __builtin_amdgcn_wmma_f32_16x16x4_f32(false, a, false, b2, (short)0, acc2, false, false);
        acc3 = __builtin_amdgcn_wmma_f32_16x16x4_f32(false, a, false, b3, (short)0, acc3, false, false);
    }

    v8f accs[4] = {acc0, acc1, acc2, acc3};
#pragma unroll
    for (int j = 0; j < 4; ++j) {
        const int   nc  = n0 + j * 16 + lid;
        const float bsv = bias ? bias[nc] : 0.0f;
#pragma unroll
        for (int r = 0; r < 8; ++r) {
            int   row = m0 + r + 8 * half;
            float v   = accs[j][r];
            if (rscale) v *= rscale[(long long)blockIdx.z * rsStride + row];
            v += bsv;
            if (act == 1) v = v > 0.0f ? v : expm1f(v);
            Cb[(long long)row * N + nc] = v;
        }
    }
}

// ---------------------------------------------------------------------------
// GCN edge aggregation: h[d,:] += dis[s]*dis[d] * xw[s,:]   (F = 256)
// ---------------------------------------------------------------------------
__global__ __launch_bounds__(256)
void gcn_aggregate(const float* __restrict__ xw, const int* __restrict__ src,
                   const int* __restrict__ dst, const float* __restrict__ dis,
                   float* __restrict__ h, int E) {
    int e = blockIdx.x;
    if (e >= E) return;
    int   s   = src[e], d = dst[e];
    float nrm = dis[s] * dis[d];
    int   f   = threadIdx.x;
    atomicAdd(&h[(long long)d * F256 + f], nrm * xw[(long long)s * F256 + f]);
}

// h[i,f] = act( h[i,f] + dis[i]^2 * xw[i,f] + b[f] )
__global__ __launch_bounds__(256)
void self_bias_act(const float* __restrict__ xw, const float* __restrict__ dis,
                   const float* __restrict__ b, float* __restrict__ h,
                   long long n, int act) {
    long long idx = (long long)blockIdx.x * 256 + threadIdx.x;
    long long st  = (long long)gridDim.x * 256;
    for (; idx < n; idx += st) {
        long long i = idx >> 8;
        int       f = idx & 255;
        float v = h[idx] + dis[i] * dis[i] * xw[idx] + b[f];
        if (act) v = v > 0.0f ? v : expm1f(v);
        h[idx] = v;
    }
}

// ---------------------------------------------------------------------------
// Instance norm (per graph, per feature) over 4096 nodes
// ---------------------------------------------------------------------------
__global__ __launch_bounds__(256)
void inorm_stats(const float* __restrict__ h, float* __restrict__ meanb,
                 float* __restrict__ rstdb) {
    __shared__ float s1[256], s2[256];
    int bx = blockIdx.x;            // g*256 + f
    int g = bx >> 8, f = bx & 255, t = threadIdx.x;
    float a = 0.0f, b = 0.0f;
    for (int i = t; i < DEV_N; i += 256) {
        float v = h[(long long)((g << 12) + i) * F256 + f];
        a += v; b += v * v;
    }
    s1[t] = a; s2[t] = b; __syncthreads();
    for (int s = 128; s > 0; s >>= 1) {
        if (t < s) { s1[t] += s1[t + s]; s2[t] += s2[t + s]; }
        __syncthreads();
    }
    if (t == 0) {
        float m   = s1[0] * (1.0f / DEV_N);
        float var = s2[0] * (1.0f / DEV_N) - m * m;
        meanb[bx] = m;
        rstdb[bx] = rsqrtf(var + 1e-5f);
    }
}
__global__ __launch_bounds__(256)
void inorm_apply(float* __restrict__ h, const float* __restrict__ meanb,
                 const float* __restrict__ rstdb) {
    long long idx = (long long)blockIdx.x * 256 + threadIdx.x;
    long long st  = (long long)gridDim.x * 256;
    long long n   = (long long)NTOT * F256;
    for (; idx < n; idx += st) {
        long long i = idx >> 8;
        int f = idx & 255, g = (int)(i >> 12);
        int s = (g << 8) + f;
        h[idx] = (h[idx] - meanb[s]) * rstdb[s];
    }
}

// ---------------------------------------------------------------------------
// avg_pool: cluster means
// ---------------------------------------------------------------------------
__global__ __launch_bounds__(256)
void pool_accum(const float* __restrict__ y, const int* __restrict__ in_batch,
                const int* __restrict__ cluster, float* __restrict__ px,
                float* __restrict__ cnt) {
    int i  = blockIdx.x;
    int gc = in_batch[i] * DEV_C + cluster[i];
    int f  = threadIdx.x;
    atomicAdd(&px[(long long)gc * F256 + f], y[(long long)i * F256 + f]);
    if (f == 0) atomicAdd(&cnt[gc], 1.0f);
}
__global__ __launch_bounds__(256)
void pool_div(float* __restrict__ px, const float* __restrict__ cnt) {
    long long idx = (long long)blockIdx.x * 256 + threadIdx.x;
    long long n   = (long long)DEV_M * F256;
    if (idx < n) px[idx] /= fmaxf(cnt[idx >> 8], 1.0f);
}

// ---------------------------------------------------------------------------
// Pooled adjacency (block-diagonal, stored transposed: AT[g][d][s])
// ---------------------------------------------------------------------------
__global__ __launch_bounds__(256)
void build_A(const int* __restrict__ src, const int* __restrict__ dst,
             const int* __restrict__ in_batch, const int* __restrict__ cluster,
             float* __restrict__ AT, int E) {
    int e = blockIdx.x * 256 + threadIdx.x;
    if (e >= E) return;
    int s = src[e], d = dst[e];
    int g = in_batch[s];
    AT[(long long)g * (DEV_C * DEV_C) + (long long)cluster[d] * DEV_C + cluster[s]] = 1.0f;
}
__global__ __launch_bounds__(256) void A_diag(float* AT) {
    int m = blockIdx.x * 256 + threadIdx.x;
    if (m < DEV_M) {
        int g = m >> 9, i = m & 511;
        AT[(long long)g * (DEV_C * DEV_C) + (long long)i * DEV_C + i] = 1.0f;
    }
}
// dic[m] = rsqrt( sum_s AT[m][s] )   (row sum of AT == column sum of Ahat)
__global__ __launch_bounds__(256) void A_coldeg(const float* __restrict__ AT,
                                                float* __restrict__ dic) {
    int m = blockIdx.x * 256 + threadIdx.x;
    if (m < DEV_M) {
        const float* row = AT + (long long)m * DEV_C;
        float s = 0.0f;
        for (int k = 0; k < DEV_C; ++k) s += row[k];
        dic[m] = rsqrtf(s);
    }
}
// pre-scale columns: AT[g][d][s] *= dic[g*512+s]
__global__ __launch_bounds__(256) void A_scale(float* __restrict__ AT,
                                               const float* __restrict__ dic) {
    long long idx = (long long)blockIdx.x * 256 + threadIdx.x;
    if (idx < (long long)DEV_B * DEV_C * DEV_C) {
        int g = (int)(idx >> 18);
        int s = (int)(idx & 511);
        AT[idx] *= dic[(g << 9) + s];
    }
}

// ---------------------------------------------------------------------------
// Final fused FC:  out = elu(Kmat @ W2 + b2) @ W3  (out pre-seeded with b3)
// Kmat [32768,512] (flat view of z), W2 [512,1024], W3 [1024,3].
// Wave tile 16x64 (4 accumulators); block tile 64x128 of elu(K@W2+b2),
// contracted with W3, shfl_xor reduction over the 16 h-lanes, 3 atomics/row.
// ---------------------------------------------------------------------------
__global__ __launch_bounds__(256)
void fc_fused(const float* __restrict__ Kmat, const float* __restrict__ W2,
              const float* __restrict__ b2, const float* __restrict__ W3,
              float* __restrict__ out) {
    const int Kdim = DEV_C;      // 512
    const int H    = 1024;
    const int lane = threadIdx.x & 31;
    const int wid  = threadIdx.x >> 5;
    const int half = lane >> 4;
    const int lid  = lane & 15;
    const int m0   = blockIdx.y * 64 + (wid & 3) * 16;
    const int h0   = blockIdx.x * 128 + (wid >> 2) * 64;

    const float* Arow = Kmat + (long long)(m0 + lid) * Kdim;
    const int    kb   = half * 2;

    v8f acc0 = {}, acc1 = {}, acc2 = {}, acc3 = {};
#pragma unroll 2
    for (int k0 = 0; k0 < Kdim; k0 += 4) {
        v2f a;
        a.x = Arow[k0 + kb];
        a.y = Arow[k0 + kb + 1];

        const float* Brow0 = W2 + (long long)(k0 + kb) * H + h0 + lid;
        const float* Brow1 = Brow0 + H;
        if (k0 + 8 < Kdim) __builtin_prefetch(Brow0 + (long long)8 * H, 0, 1);

        v2f b0, b1, b2, b3;
        b0.x = Brow0[0];  b0.y = Brow1[0];
        b1.x = Brow0[16]; b1.y = Brow1[16];
        b2.x = Brow0[32]; b2.y = Brow1[32];
        b3.x = Brow0[48]; b3.y = Brow1[48];

        acc0 = __builtin_amdgcn_wmma_f32_16x16x4_f32(false, a, false, b0, (short)0, acc0, false, false);
        acc1 = __builtin_amdgcn_wmma_f32_16x16x4_f32(false, a, false, b1, (short)0, acc1, false, false);
        acc2 = __builtin_amdgcn_wmma_f32_16x16x4_f32(false, a, false, b2, (short)0, acc2, false, false);
        acc3 = __builtin_amdgcn_wmma_f32_16x16x4_f32(false, a, false, b3, (short)0, acc3, false, false);
    }

    v8f accs[4] = {acc0, acc1, acc2, acc3};
#pragma unroll
    for (int j = 0; j < 4; ++j) {
        const int   h   = h0 + j * 16 + lid;
        const float bb  = b2[h];
        const float w30 = W3[h * 3 + 0];
        const float w31 = W3[h * 3 + 1];
        const float w32 = W3[h * 3 + 2];
#pragma unroll
        for (int r = 0; r < 8; ++r) {
            int   row = m0 + r + 8 * half;
            float v   = accs[j][r] + bb;
            v = v > 0.0f ? v : expm1f(v);
            float p0 = v * w30, p1 = v * w31, p2 = v * w32;
#pragma unroll
            for (int msk = 1; msk < 16; msk <<= 1) {   // reduce over lid (same row)
                p0 += __shfl_xor(p0, msk, 32);
                p1 += __shfl_xor(p1, msk, 32);
                p2 += __shfl_xor(p2, msk, 32);
            }
            if (lid == 0) {
                atomicAdd(&out[(long long)row * 3 + 0], p0);
                atomicAdd(&out[(long long)row * 3 + 1], p1);
                atomicAdd(&out[(long long)row * 3 + 2], p2);
            }
        }
    }
}

__global__ __launch_bounds__(256) void out_init(float* out, const float* b3) {
    int i = blockIdx.x * 256 + threadIdx.x;
    if (i < NTOT * 3) out[i] = b3[i % 3];
}

// ---------------------------------------------------------------------------
// Host orchestration
// ---------------------------------------------------------------------------
extern "C" void kernel_launch(void* const* d_in, const int* in_sizes, int n_in,
                              void* d_out, int out_size, void* d_ws, size_t ws_size,
                              hipStream_t stream) {
    const float* x    = (const float*)d_in[0];
    const float* w_g1 = (const float*)d_in[1];  const float* b_g1 = (const float*)d_in[2];
    const float* w_g2 = (const float*)d_in[3];  const float* b_g2 = (const float*)d_in[4];
    const float* w_g3 = (const float*)d_in[5];  const float* b_g3 = (const float*)d_in[6];
    const float* w_t1 = (const float*)d_in[7];  const float* b_t1 = (const float*)d_in[8];
    const float* w_t2 = (const float*)d_in[9];  const float* b_t2 = (const float*)d_in[10];
    const float* w_g4 = (const float*)d_in[11]; const float* b_g4 = (const float*)d_in[12];
    const float* w_g5 = (const float*)d_in[13]; const float* b_g5 = (const float*)d_in[14];
    const float* w_f1 = (const float*)d_in[15]; const float* b_f1 = (const float*)d_in[16];
    const float* w_f2 = (const float*)d_in[17]; const float* b_f2 = (const float*)d_in[18];
    const float* w_f3 = (const float*)d_in[19]; const float* b_f3 = (const float*)d_in[20];
    const int*   adj  = (const int*)d_in[21];
    const int*   in_batch = (const int*)d_in[23];
    const int*   cluster  = (const int*)d_in[24];
    float* out = (float*)d_out;

    const int E   = in_sizes[21] / 2;           // 524288
    const int* src = adj;
    const int* dst = adj + E;

    // ---- scratch layout (floats) ----
    float* W = (float*)d_ws;
    float* R0   = W;                            // 16,777,216 f: ping/pong, later z
    float* ping = R0;
    float* pong = R0 + (long long)NTOT * F256;  // +8,388,608
    size_t off = 16777216;
    float* dis   = W + off; off += NTOT;        // 32768
    float* meanb = W + off; off += DEV_B * F256;
    float* rstdb = W + off; off += DEV_B * F256;
    float* px    = W + off; off += (long long)DEV_M * F256;   // 1,048,576
    float* cnt   = W + off; off += DEV_M;
    float* z1    = W + off; off += (long long)DEV_M * 32;
    float* z2    = W + off; off += (long long)DEV_M * 128;
    float* zw    = W + off; off += (long long)DEV_M * F256;
    float* z3    = W + off; off += (long long)DEV_M * F256;
    float* z4    = W + off; off += (long long)DEV_M * F256;
    float* AT    = W + off; off += (long long)DEV_B * DEV_C * DEV_C;  // 2,097,152
    float* dic   = W + off; off += DEV_M;

    const long long nNF = (long long)NTOT * F256;   // 8,388,608
    const dim3 blk(256);

    // ---- degrees ----
    fill_f32<<<128, blk, 0, stream>>>(dis, 0.0f, NTOT);
    deg_accum<<<(E + 255) / 256, blk, 0, stream>>>(dst, dis, E);
    deg_final<<<(NTOT + 255) / 256, blk, 0, stream>>>(dis, NTOT);

    // ---- GCN layer 1: xw = x@w_g1 ; aggregate ; +self +bias, ELU ----
    wmma_gemm_w<<<dim3(F256 / 128, NTOT / 64, 1), blk, 0, stream>>>(
        x, w_g1, nullptr, nullptr, ping, NTOT, F256, 64, 0, 0, 0, 0, 0);
    fill_f32<<<4096, blk, 0, stream>>>(pong, 0.0f, nNF);
    gcn_aggregate<<<E, blk, 0, stream>>>(ping, src, dst, dis, pong, E);
    self_bias_act<<<8192, blk, 0, stream>>>(ping, dis, b_g1, pong, nNF, 1);

    // ---- GCN layer 2 ----
    wmma_gemm_w<<<dim3(F256 / 128, NTOT / 64, 1), blk, 0, stream>>>(
        pong, w_g2, nullptr, nullptr, ping, NTOT, F256, F256, 0, 0, 0, 0, 0);
    fill_f32<<<4096, blk, 0, stream>>>(pong, 0.0f, nNF);
    gcn_aggregate<<<E, blk, 0, stream>>>(ping, src, dst, dis, pong, E);
    self_bias_act<<<8192, blk, 0, stream>>>(ping, dis, b_g2, pong, nNF, 1);

    // ---- GCN layer 3 (no ELU) ----
    wmma_gemm_w<<<dim3(F256 / 128, NTOT / 64, 1), blk, 0, stream>>>(
        pong, w_g3, nullptr, nullptr, ping, NTOT, F256, F256, 0, 0, 0, 0, 0);
    fill_f32<<<4096, blk, 0, stream>>>(pong, 0.0f, nNF);
    gcn_aggregate<<<E, blk, 0, stream>>>(ping, src, dst, dis, pong, E);
    self_bias_act<<<8192, blk, 0, stream>>>(ping, dis, b_g3, pong, nNF, 0);

    // ---- instance norm (in place on pong) ----
    inorm_stats<<<DEV_B * F256, blk, 0, stream>>>(pong, meanb, rstdb);
    inorm_apply<<<8192, blk, 0, stream>>>(pong, meanb, rstdb);

    // ---- avg_pool ----
    fill_f32<<<1024, blk, 0, stream>>>(px, 0.0f, (long long)DEV_M * F256);
    fill_f32<<<16, blk, 0, stream>>>(cnt, 0.0f, DEV_M);
    pool_accum<<<NTOT, blk, 0, stream>>>(pong, in_batch, cluster, px, cnt);
    pool_div<<<(DEV_M * F256 + 255) / 256, blk, 0, stream>>>(px, cnt);

    // ---- small MLP on pooled features ----
    wmma_gemm<<<dim3(1, DEV_M / 64, 1), blk, 0, stream>>>(         // 4096x256 @ 256x32
        px, w_t1, b_t1, z1, DEV_M, 32, F256, 1);
    wmma_gemm_w<<<dim3(1, DEV_M / 64, 1), blk, 0, stream>>>(       // 4096x32 @ 32x128
        z1, w_t2, b_t2, nullptr, z2, DEV_M, 128, 32, 1, 0, 0, 0, 0);

    // ---- pooled adjacency (transposed, block-diagonal) ----
    const long long nA = (long long)DEV_B * DEV_C * DEV_C;
    fill_f32<<<2048, blk, 0, stream>>>(AT, 0.0f, nA);
    build_A<<<(E + 255) / 256, blk, 0, stream>>>(src, dst, in_batch, cluster, AT, E);
    A_diag<<<16, blk, 0, stream>>>(AT);
    A_coldeg<<<16, blk, 0, stream>>>(AT, dic);
    A_scale<<<(int)((nA + 255) / 256), blk, 0, stream>>>(AT, dic);

    // ---- pooled GCN layer 4:  z3 = elu(dic * (AT @ (z2@w_g4)) + b_g4) ----
    wmma_gemm_w<<<dim3(F256 / 128, DEV_M / 64, 1), blk, 0, stream>>>(
        z2, w_g4, nullptr, nullptr, zw, DEV_M, F256, 128, 0, 0, 0, 0, 0);
    wmma_gemm_w<<<dim3(F256 / 128, DEV_C / 64, DEV_B), blk, 0, stream>>>(
        AT, zw, b_g4, dic, z3, DEV_C, F256, DEV_C, 1,
        (long long)DEV_C * DEV_C, (long long)DEV_C * F256,
        (long long)DEV_C * F256, DEV_C);

    // ---- pooled GCN layer 5 ----
    wmma_gemm_w<<<dim3(F256 / 128, DEV_M / 64, 1), blk, 0, stream>>>(
        z3, w_g5, nullptr, nullptr, zw, DEV_M, F256, F256, 0, 0, 0, 0, 0);
    wmma_gemm_w<<<dim3(F256 / 128, DEV_C / 64, DEV_B), blk, 0, stream>>>(
        AT, zw, b_g5, dic, z4, DEV_C, F256, DEV_C, 1,
        (long long)DEV_C * DEV_C, (long long)DEV_C * F256,
        (long long)DEV_C * F256, DEV_C);

    // ---- z = z4 @ w_f1 + b_f1  -> [4096,4096], reuses R0 (flat = k [32768,512]) ----
    wmma_gemm_w<<<dim3(DEV_N / 128, DEV_M / 64, 1), blk, 0, stream>>>(
        z4, w_f1, b_f1, nullptr, R0, DEV_M, DEV_N, F256, 0, 0, 0, 0, 0);

    // ---- fused final FCs: out = elu(k@w_f2+b_f2)@w_f3 + b_f3 ----
    out_init<<<(NTOT * 3 + 255) / 256, blk, 0, stream>>>(out, b_f3);
    fc_fused<<<dim3(1024 / 128, NTOT / 64, 1), blk, 0, stream>>>(
        R0, w_f2, b_f2, w_f3, out);
}